// LightGCNConv_27642409517734
// MI455X (gfx1250) — compile-verified
//
#include <hip/hip_runtime.h>

#ifndef __has_builtin
#define __has_builtin(x) 0
#endif

// ---------------------------------------------------------------------------
// CDNA5 async global->LDS staging (ASYNCcnt path). Guarded: if the builtin is
// absent we fall back to a plain (correct) synchronous copy.
// ---------------------------------------------------------------------------
#if __has_builtin(__builtin_amdgcn_global_load_async_to_lds_b32)
#define HAVE_ASYNC_LDS 1
#else
#define HAVE_ASYNC_LDS 0
#endif

typedef __attribute__((address_space(1))) int gas_int;  // global int
typedef __attribute__((address_space(3))) int las_int;  // LDS int

__device__ __forceinline__ void async_cp_b32(void* lds_dst, const void* g_src) {
#if HAVE_ASYNC_LDS
  __builtin_amdgcn_global_load_async_to_lds_b32(
      (gas_int*)g_src, (las_int*)lds_dst, /*offset=*/0, /*cpol=*/0);
#else
  *(volatile int*)lds_dst = *(const volatile int*)g_src;
#endif
}

// s_wait_asynccnt <= N (wave-local). Inline asm: works on both toolchains and
// doubles as a compiler memory barrier for the LDS data written by the async
// engine. If HAVE_ASYNC_LDS==0 the counter is always 0 and this is a no-op.
#define WAIT_ASYNC(n) asm volatile("s_wait_asynccnt %0" ::"n"(n) : "memory")

__device__ __forceinline__ void atom_add_f32(float* p, float v) {
  // Non-returning hardware global_atomic_add_f32, device (agent) scope.
  (void)__hip_atomic_fetch_add(p, v, __ATOMIC_RELAXED, __HIP_MEMORY_SCOPE_AGENT);
}

#define DIM 64
#define WPB 8   // waves per block (256 threads)
#define EPS 32  // edges per slice (one per lane during staging)

// Process one edge: 16 lanes cooperate, each covers float4 `el` of the row.
__device__ __forceinline__ void process_edge(
    const int* __restrict__ sr, const int* __restrict__ sc,
    const float* __restrict__ sv, int idx, int el,
    const float* __restrict__ user_emb, const float* __restrict__ item_emb,
    float* __restrict__ out_user, float* __restrict__ out_item) {
  const int   r = sr[idx];  // LDS broadcast across the 16-lane group
  const int   c = sc[idx];
  const float v = sv[idx];

  const float4 it = *reinterpret_cast<const float4*>(item_emb + c * DIM + el);
  const float4 us = *reinterpret_cast<const float4*>(user_emb + r * DIM + el);

  float* pu = out_user + r * DIM + el;
  float* pi = out_item + c * DIM + el;
  atom_add_f32(pu + 0, v * it.x);
  atom_add_f32(pu + 1, v * it.y);
  atom_add_f32(pu + 2, v * it.z);
  atom_add_f32(pu + 3, v * it.w);
  atom_add_f32(pi + 0, v * us.x);
  atom_add_f32(pi + 1, v * us.y);
  atom_add_f32(pi + 2, v * us.z);
  atom_add_f32(pi + 3, v * us.w);
}

// ---------------------------------------------------------------------------
// COO scatter for both directions in one pass (emitted FIRST in the module so
// the disasm snippet shows it):
//   user_new[r] += v * item_emb[c]
//   item_new[c] += v * user_emb[r]
// 256 threads = 8 waves. Each wave owns 32-edge slices; the edge stream
// (rows/cols/vals) is async-staged into a wave-private double buffer in LDS
// (no barriers needed). Processing: 16 lanes x float4 = one 64-float row;
// 2 edges in flight per wave pass.
// ---------------------------------------------------------------------------
__global__ void __launch_bounds__(256)
lightgcn_scatter(const float* __restrict__ user_emb,
                 const float* __restrict__ item_emb,
                 const int* __restrict__ rows,
                 const int* __restrict__ cols,
                 const float* __restrict__ vals,
                 float* __restrict__ out_user,
                 float* __restrict__ out_item,
                 int nnz) {
  __shared__ int   s_rows[WPB][2][EPS];
  __shared__ int   s_cols[WPB][2][EPS];
  __shared__ float s_vals[WPB][2][EPS];

  const int lane = (int)(threadIdx.x & 31);
  const int wid  = (int)(threadIdx.x >> 5);
  const int nSlices     = (nnz + EPS - 1) / EPS;
  const int sliceStride = (int)gridDim.x * WPB;

  int s = (int)blockIdx.x * WPB + wid;
  if (s >= nSlices) return;  // uniform per wave

  // Prologue: stage first slice into buffer 0 (3 async b32 ops per wave).
  {
    const int e = s * EPS + lane;
    if (e < nnz) {
      async_cp_b32(&s_rows[wid][0][lane], rows + e);
      async_cp_b32(&s_cols[wid][0][lane], cols + e);
      async_cp_b32(&s_vals[wid][0][lane], vals + e);
    }
  }

  const int g  = lane >> 4;         // edge group within wave: 0 or 1
  const int el = (lane & 15) << 2;  // float4 element offset within row

  int buf = 0;
  for (; s < nSlices; s += sliceStride) {
    const int snext = s + sliceStride;
    if (snext < nSlices) {
      // Prefetch next slice into the other buffer, then wait only for the
      // current slice's 3 async ops (in-order completion => <=3 suffices).
      const int e = snext * EPS + lane;
      if (e < nnz) {
        async_cp_b32(&s_rows[wid][buf ^ 1][lane], rows + e);
        async_cp_b32(&s_cols[wid][buf ^ 1][lane], cols + e);
        async_cp_b32(&s_vals[wid][buf ^ 1][lane], vals + e);
      }
      WAIT_ASYNC(3);
    } else {
      WAIT_ASYNC(0);
    }

    const int* sr = s_rows[wid][buf];
    const int* sc = s_cols[wid][buf];
    const float* sv = s_vals[wid][buf];
    const int base = s * EPS;

    if (base + EPS <= nnz) {
      // Fast path: full slice, no bounds checks in the hot loop.
#pragma unroll 4
      for (int p = 0; p < EPS / 2; ++p)
        process_edge(sr, sc, sv, p * 2 + g, el, user_emb, item_emb, out_user, out_item);
    } else {
      // Tail slice (at most one per wave).
      for (int p = 0; p < EPS / 2; ++p) {
        const int idx = p * 2 + g;
        if (base + idx < nnz)
          process_edge(sr, sc, sv, idx, el, user_emb, item_emb, out_user, out_item);
      }
    }
    buf ^= 1;
  }
}

// ---------------------------------------------------------------------------
// Zero the (poisoned / previously accumulated) output buffer.
// ---------------------------------------------------------------------------
__global__ void __launch_bounds__(256)
lightgcn_zero(float* __restrict__ out, int n) {
  const int stride = (int)(gridDim.x * blockDim.x);
  const int tid    = (int)(blockIdx.x * blockDim.x + threadIdx.x);
  const int n4     = n >> 2;
  float4* __restrict__ out4 = reinterpret_cast<float4*>(out);
  for (int i = tid; i < n4; i += stride)
    out4[i] = make_float4(0.f, 0.f, 0.f, 0.f);
  const int base = n4 << 2;
  if (tid < (n & 3)) out[base + tid] = 0.f;
}

// ---------------------------------------------------------------------------
extern "C" void kernel_launch(void* const* d_in, const int* in_sizes, int n_in,
                              void* d_out, int out_size, void* d_ws, size_t ws_size,
                              hipStream_t stream) {
  (void)n_in; (void)d_ws; (void)ws_size;

  const float* user_emb = (const float*)d_in[0];
  const float* item_emb = (const float*)d_in[1];
  const int*   rows     = (const int*)d_in[2];
  const int*   cols     = (const int*)d_in[3];
  const float* vals     = (const float*)d_in[4];

  const int n_users = in_sizes[0] / DIM;  // 100000
  const int nnz     = in_sizes[2];        // 10000000

  float* out      = (float*)d_out;
  float* out_user = out;
  float* out_item = out + (size_t)n_users * DIM;

  // Zero-fill outputs (outputs are accumulated via atomics).
  int zgrid = (out_size / 4 + 255) / 256;
  if (zgrid > 4096) zgrid = 4096;
  if (zgrid < 1) zgrid = 1;
  lightgcn_zero<<<zgrid, 256, 0, stream>>>(out, out_size);

  // Scatter pass.
  const int nSlices = (nnz + EPS - 1) / EPS;
  int sgrid = (nSlices + WPB - 1) / WPB;
  if (sgrid > 4096) sgrid = 4096;  // grid-stride => >=2 slices/wave for pipelining
  if (sgrid < 1) sgrid = 1;
  lightgcn_scatter<<<sgrid, 256, 0, stream>>>(user_emb, item_emb, rows, cols, vals,
                                              out_user, out_item, nnz);
}